// GraphPool_36069135352326
// MI455X (gfx1250) — compile-verified
//
#include <hip/hip_runtime.h>
#include <hip/hip_bf16.h>

typedef __attribute__((ext_vector_type(2))) float v2f;
typedef __attribute__((ext_vector_type(8))) float v8f;

#define FDIM   512
#define NNODES 4096
#define BATCH  16
#define NKEEP  2048

// ---------------------------------------------------------------------------
// Kernel 1: scores[b*N+n] = sigmoid(dot(h[b,n,:], W) + bias)
// Per wave: 16 rows of h. A operand = 16x4 f32 tile (lane l<16: K={0,1},
// lane l+16: K={2,3}, row = lane&15). B operand = W[k..k+3] replicated to
// all 16 columns, so every column of the 16x16 C tile holds the 16 dots.
// Emits v_wmma_f32_16x16x4_f32 (XDL f32 path, exact f32 math).
// ---------------------------------------------------------------------------
__global__ __launch_bounds__(256) void gp_score_kernel(
    const float* __restrict__ h, const float* __restrict__ W,
    const float* __restrict__ bias, float* __restrict__ scores) {
  const int lane    = threadIdx.x & 31;
  const int wave    = blockIdx.x * (blockDim.x >> 5) + (threadIdx.x >> 5);
  const int rowBase = wave * 16;                 // 16 rows of [B*N, F] per wave
  const int m       = lane & 15;                 // row within tile
  const int half    = lane >> 4;                 // K sub-offset selector

  const float* hrow = h + (size_t)(rowBase + m) * FDIM + 2 * half;
  const float* wp   = W + 2 * half;

  v8f c = {};
#pragma unroll 8
  for (int k = 0; k < FDIM; k += 4) {
    v2f a = *(const v2f*)(hrow + k);             // A[m][k+2h .. k+2h+1]
    v2f w = *(const v2f*)(wp + k);               // B[k+2h .. k+2h+1][*] (bcast)
    c = __builtin_amdgcn_wmma_f32_16x16x4_f32(false, a, false, w,
                                              (short)0, c, false, false);
  }

  // C layout: VGPR r, lanes 0-15 -> row M=r; lanes 16-31 -> row M=8+r.
  // All columns identical (B replicated): lane 0 stores rows 0..7,
  // lane 16 stores rows 8..15.
  if ((lane & 15) == 0) {
    const float bv = bias[0];
    const int off  = rowBase + half * 8;
#pragma unroll
    for (int r = 0; r < 8; ++r) {
      float x = c[r] + bv;
      scores[off + r] = 1.0f / (1.0f + __expf(-x));
    }
  }
}

// ---------------------------------------------------------------------------
// Kernel 2: per-batch exact top-k ordering via in-LDS bitonic sort of
// 64-bit keys (score_bits << 32) | (~index). Scores are sigmoid outputs in
// (0,1): positive floats, so the raw bit pattern is order-monotone. The
// ~index low word makes ties resolve to the smaller index first, matching
// lax.top_k's stable descending order. 4096 keys = 32 KB LDS.
// ---------------------------------------------------------------------------
__global__ __launch_bounds__(1024) void gp_topk_kernel(
    const float* __restrict__ scores, unsigned int* __restrict__ idx_out) {
  __shared__ unsigned long long key[NNODES];
  const int b   = blockIdx.x;
  const int tid = threadIdx.x;

  for (int i = tid; i < NNODES; i += 1024) {
    unsigned int bits = __float_as_uint(scores[b * NNODES + i]);
    key[i] = ((unsigned long long)bits << 32) |
             (unsigned long long)(0xFFFFFFFFu - (unsigned int)i);
  }
  __syncthreads();

  for (int k = 2; k <= NNODES; k <<= 1) {
    for (int j = k >> 1; j > 0; j >>= 1) {
#pragma unroll
      for (int t = 0; t < 4; ++t) {
        const int i   = tid + t * 1024;
        const int ixj = i ^ j;
        if (ixj > i) {                            // one owner per pair
          unsigned long long a  = key[i];
          unsigned long long cc = key[ixj];
          const bool desc = ((i & k) == 0);
          if (desc ? (a < cc) : (a > cc)) { key[i] = cc; key[ixj] = a; }
        }
      }
      __syncthreads();
    }
  }

  for (int j = tid; j < NKEEP; j += 1024) {
    idx_out[b * NKEEP + j] =
        0xFFFFFFFFu - (unsigned int)(key[j] & 0xFFFFFFFFull);
  }
}

// ---------------------------------------------------------------------------
// Kernel 3: out[b, j, :] = h[b, idx[b,j], :] * score[b, idx[b,j]]
// One 128-thread block per kept row; one float4 per thread (512 floats).
// ---------------------------------------------------------------------------
__global__ __launch_bounds__(128) void gp_gather_kernel(
    const float* __restrict__ h, const float* __restrict__ scores,
    const unsigned int* __restrict__ idx, float* __restrict__ out) {
  const int row = blockIdx.x;                     // 0 .. B*NKEEP-1
  const int b   = row / NKEEP;
  const int j   = row - b * NKEEP;
  const unsigned int n = idx[b * NKEEP + j];
  const float s = scores[b * NNODES + n];

  const float4* src = (const float4*)(h + ((size_t)b * NNODES + n) * FDIM);
  float4*       dst = (float4*)(out + (size_t)row * FDIM);

  float4 v = src[threadIdx.x];
  v.x *= s; v.y *= s; v.z *= s; v.w *= s;
  dst[threadIdx.x] = v;
}

// ---------------------------------------------------------------------------
extern "C" void kernel_launch(void* const* d_in, const int* in_sizes, int n_in,
                              void* d_out, int out_size, void* d_ws,
                              size_t ws_size, hipStream_t stream) {
  const float* h    = (const float*)d_in[0];   // [16, 4096, 512] f32
  const float* W    = (const float*)d_in[1];   // [512, 1] f32
  const float* bias = (const float*)d_in[2];   // [1] f32
  float*       out  = (float*)d_out;           // [16, 2048, 512] f32

  float*        scores = (float*)d_ws;                               // 256 KB
  unsigned int* idx    = (unsigned int*)((char*)d_ws +
                          (size_t)BATCH * NNODES * sizeof(float));   // 128 KB

  // 65536 rows / 16 rows-per-wave / 8 waves-per-block = 512 blocks
  gp_score_kernel<<<(BATCH * NNODES) / (16 * 8), 256, 0, stream>>>(
      h, W, bias, scores);
  gp_topk_kernel<<<BATCH, 1024, 0, stream>>>(scores, idx);
  gp_gather_kernel<<<BATCH * NKEEP, 128, 0, stream>>>(h, scores, idx, out);
}